// InteractiveBlock_87436944212530
// MI455X (gfx1250) — compile-verified
//
#include <hip/hip_runtime.h>
#include <hip/hip_bf16.h>

typedef __attribute__((ext_vector_type(2))) float v2f;
typedef __attribute__((ext_vector_type(8))) float v8f;

#define B_    32
#define C_    256
#define CE_   128
#define H_    56
#define W_    56
#define HE_   28
#define WE_   28
#define PIX_  (H_*W_)     // 3136
#define PIXE_ (HE_*WE_)   // 784

// ---------------- Kernel 0: collapse conv weights to a single 128-vector ----
__global__ void prep_weights(const float* __restrict__ conv_w,  // [256,128]
                             const float* __restrict__ conv_b,  // [256]
                             float* __restrict__ wm) {          // [129]
  int tid = threadIdx.x;
  if (tid < CE_) {
    float s = 0.f;
#pragma unroll 8
    for (int o = 0; o < C_; ++o) s += conv_w[o * CE_ + tid];
    wm[tid] = s * (1.0f / C_);
  } else if (tid == CE_) {
    float s = 0.f;
    for (int o = 0; o < C_; ++o) s += conv_b[o];
    wm[CE_] = s * (1.0f / C_);
  }
}

// ---------------- Kernel A: per-sample mask (one block per sample) ----------
__global__ void __launch_bounds__(256)
mask_kernel(const float* __restrict__ emb,    // [32,128,28,28]
            const float* __restrict__ infx,   // [32,1,112,112]
            const float* __restrict__ wm,     // [129] = w_mean, b_mean
            float* __restrict__ mask_out) {   // [32,3136]
  __shared__ float s_wm[CE_];
  __shared__ float s_small[PIXE_];
  __shared__ float s_m[PIX_];
  __shared__ float s_rmin[256];
  __shared__ float s_rmax[256];
  __shared__ float s_bm;

  const int n = blockIdx.x;
  const int tid = threadIdx.x;
  if (tid < CE_) s_wm[tid] = wm[tid];
  if (tid == CE_) s_bm = wm[CE_];
  __syncthreads();
  const float bm = s_bm;

  // -- Step 1: small[p] = sum_c w_mean[c]*E[n,c,p] + b_mean, via WMMA f32 16x16x4.
  // A: 16 pixels (M) x 4 channels (K); B: w_mean broadcast across all 16 columns.
  // Per ISA VGPR layout: A/B VGPR0 holds K0 (lanes 0-15) / K2 (lanes 16-31),
  // VGPR1 holds K1 / K3. Every column of D is identical.
  {
    const int wave = tid >> 5;
    const int lane = tid & 31;
    const bool hi = lane >= 16;
    const int m = hi ? lane - 16 : lane;
    const float* En = emb + (size_t)n * CE_ * PIXE_;
    for (int t = wave; t < PIXE_ / 16; t += 8) {   // 49 tiles of 16 pixels
      const int p0 = t * 16;
      const int pix = p0 + m;
      v8f acc = {};
#pragma unroll
      for (int kb = 0; kb < CE_; kb += 4) {
        const int ka = kb + (hi ? 2 : 0);
        v2f A, Bv;
        A.x  = En[ka * PIXE_ + pix];
        A.y  = En[(ka + 1) * PIXE_ + pix];
        Bv.x = s_wm[ka];
        Bv.y = s_wm[ka + 1];
        acc = __builtin_amdgcn_wmma_f32_16x16x4_f32(false, A, false, Bv,
                                                    (short)0, acc, false, false);
      }
      // D layout: VGPR j = row M=j (lanes 0-15) / M=j+8 (lanes 16-31); all N equal.
      if (lane == 0) {
#pragma unroll
        for (int j = 0; j < 8; ++j) s_small[p0 + j] = acc[j] + bm;
      } else if (lane == 16) {
#pragma unroll
        for (int j = 0; j < 8; ++j) s_small[p0 + 8 + j] = acc[j] + bm;
      }
    }
  }
  __syncthreads();

  // -- Step 2: bilinear 28 -> 56, align_corners=False
  for (int idx = tid; idx < PIX_; idx += 256) {
    int oy = idx / W_, ox = idx % W_;
    float sy = fminf(fmaxf(0.5f * oy - 0.25f, 0.f), (float)(HE_ - 1));
    float sx = fminf(fmaxf(0.5f * ox - 0.25f, 0.f), (float)(WE_ - 1));
    int y0 = (int)sy, x0 = (int)sx;
    int y1 = min(y0 + 1, HE_ - 1), x1 = min(x0 + 1, WE_ - 1);
    float wy = sy - y0, wx = sx - x0;
    float r0 = s_small[y0 * WE_ + x0] * (1.f - wx) + s_small[y0 * WE_ + x1] * wx;
    float r1 = s_small[y1 * WE_ + x0] * (1.f - wx) + s_small[y1 * WE_ + x1] * wx;
    s_m[idx] = r0 * (1.f - wy) + r1 * wy;
  }
  __syncthreads();

  // -- min/max #1
  float lmin = 3.402823e38f, lmax = -3.402823e38f;
  for (int idx = tid; idx < PIX_; idx += 256) {
    float v = s_m[idx]; lmin = fminf(lmin, v); lmax = fmaxf(lmax, v);
  }
  s_rmin[tid] = lmin; s_rmax[tid] = lmax; __syncthreads();
  for (int s = 128; s > 0; s >>= 1) {
    if (tid < s) {
      s_rmin[tid] = fminf(s_rmin[tid], s_rmin[tid + s]);
      s_rmax[tid] = fmaxf(s_rmax[tid], s_rmax[tid + s]);
    }
    __syncthreads();
  }
  float mn = s_rmin[0], mx = s_rmax[0];
  float inv = 1.0f / (mx - mn);
  __syncthreads();

  // -- Step 3: normalize, multiply by bilinear(inference 112 -> 56, align_corners=True)
  const float* In = infx + (size_t)n * 112 * 112;
  const float sc = 111.0f / 55.0f;
  for (int idx = tid; idx < PIX_; idx += 256) {
    int oy = idx / W_, ox = idx % W_;
    float sy = oy * sc, sx = ox * sc;
    int y0 = (int)sy, x0 = (int)sx;
    int y1 = min(y0 + 1, 111), x1 = min(x0 + 1, 111);
    float wy = sy - y0, wx = sx - x0;
    float r0 = In[y0 * 112 + x0] * (1.f - wx) + In[y0 * 112 + x1] * wx;
    float r1 = In[y1 * 112 + x0] * (1.f - wx) + In[y1 * 112 + x1] * wx;
    float iv = r0 * (1.f - wy) + r1 * wy;
    s_m[idx] = ((s_m[idx] - mn) * inv) * iv;
  }
  __syncthreads();

  // -- min/max #2 + write mask
  lmin = 3.402823e38f; lmax = -3.402823e38f;
  for (int idx = tid; idx < PIX_; idx += 256) {
    float v = s_m[idx]; lmin = fminf(lmin, v); lmax = fmaxf(lmax, v);
  }
  s_rmin[tid] = lmin; s_rmax[tid] = lmax; __syncthreads();
  for (int s = 128; s > 0; s >>= 1) {
    if (tid < s) {
      s_rmin[tid] = fminf(s_rmin[tid], s_rmin[tid + s]);
      s_rmax[tid] = fmaxf(s_rmax[tid], s_rmax[tid + s]);
    }
    __syncthreads();
  }
  mn = s_rmin[0]; mx = s_rmax[0];
  inv = 1.0f / (mx - mn);
  float* mout = mask_out + (size_t)n * PIX_;
  for (int idx = tid; idx < PIX_; idx += 256) mout[idx] = (s_m[idx] - mn) * inv;
}

// ---------------- Kernel B: bandwidth-bound elementwise (float4) ------------
// out = sigmoid(x) * (mask*w + 1). 3136 % 4 == 0 so a float4 never crosses a
// (n,channel) row and the matching mask float4 is aligned.
__global__ void __launch_bounds__(256)
ew_kernel(const float* __restrict__ xin,
          const float* __restrict__ mask,   // [32,3136], L2-resident
          const float* __restrict__ wt,
          float* __restrict__ out) {
  const float w = wt[0];
  const size_t idx = (size_t)blockIdx.x * blockDim.x + threadIdx.x;   // float4 index
  const size_t base = idx * 4;
  const size_t n   = base / (size_t)(C_ * PIX_);
  const size_t pix = base % (size_t)PIX_;

  const float4 x  = ((const float4*)xin)[idx];
  const float4 mk = *(const float4*)(mask + n * PIX_ + pix);
  float4 r;
  float a;
  a = 1.0f / (1.0f + __expf(-x.x)); r.x = a * fmaf(mk.x, w, 1.0f);
  a = 1.0f / (1.0f + __expf(-x.y)); r.y = a * fmaf(mk.y, w, 1.0f);
  a = 1.0f / (1.0f + __expf(-x.z)); r.z = a * fmaf(mk.z, w, 1.0f);
  a = 1.0f / (1.0f + __expf(-x.w)); r.w = a * fmaf(mk.w, w, 1.0f);
  ((float4*)out)[idx] = r;
}

extern "C" void kernel_launch(void* const* d_in, const int* in_sizes, int n_in,
                              void* d_out, int out_size, void* d_ws, size_t ws_size,
                              hipStream_t stream) {
  const float* main_in = (const float*)d_in[0];  // [32,256,56,56]
  const float* emb_in  = (const float*)d_in[1];  // [32,128,28,28]
  const float* inf_in  = (const float*)d_in[2];  // [32,1,112,112]
  const float* conv_w  = (const float*)d_in[3];  // [256,128]
  const float* conv_b  = (const float*)d_in[4];  // [256]
  const float* weight  = (const float*)d_in[5];  // [1]
  float* out  = (float*)d_out;

  float* wm   = (float*)d_ws;        // 129 floats (padded to 256)
  float* mask = (float*)d_ws + 256;  // 32*3136 floats

  prep_weights<<<1, 256, 0, stream>>>(conv_w, conv_b, wm);
  mask_kernel<<<B_, 256, 0, stream>>>(emb_in, inf_in, wm, mask);

  const int total4 = (B_ * C_ * PIX_) / 4;          // 6,422,528
  ew_kernel<<<total4 / 256, 256, 0, stream>>>(main_in, mask, weight, out);
}